// HCCFModel_31628139168303
// MI455X (gfx1250) — compile-verified
//
#include <hip/hip_runtime.h>

// ---------------------------------------------------------------------------
// HCCF forward on MI455X (gfx1250, wave32).
// All dense GEMMs use V_WMMA_F32_16X16X4_F32 (full fp32 precision, matching
// the fp32 reference). D = H = 128 fixed.
// ---------------------------------------------------------------------------

typedef __attribute__((ext_vector_type(2))) float v2f;
typedef __attribute__((ext_vector_type(8))) float v8f;

#define DD 128

__device__ __forceinline__ float lrelu(float x) { return x >= 0.0f ? x : 0.5f * x; }

__device__ __forceinline__ v8f wmma4(v2f a, v2f b, v8f c) {
  // D = A(16x4,f32) * B(4x16,f32) + C(16x16,f32)
  return __builtin_amdgcn_wmma_f32_16x16x4_f32(false, a, false, b, (short)0, c,
                                               false, false);
}

// ---------------------------------------------------------------------------
// prev = concat(u, i); embeds_acc = prev
// ---------------------------------------------------------------------------
__global__ __launch_bounds__(256) void k_init(const float* __restrict__ uE,
                                              const float* __restrict__ iE,
                                              float* __restrict__ prev,
                                              float* __restrict__ acc,
                                              int nu, long tot) {
  long idx = (long)blockIdx.x * blockDim.x + threadIdx.x;
  if (idx >= tot) return;
  long uElems = (long)nu * DD;
  float v = (idx < uElems) ? uE[idx] : iE[idx - uElems];
  prev[idx] = v;
  acc[idx] = v;
}

__global__ __launch_bounds__(256) void k_fill(float* __restrict__ p, long n) {
  long i = (long)blockIdx.x * blockDim.x + threadIdx.x;
  if (i < n) p[i] = 0.0f;
}

__global__ __launch_bounds__(256) void k_leaky_ip(float* __restrict__ p, long n) {
  long i = (long)blockIdx.x * blockDim.x + threadIdx.x;
  if (i < n) p[i] = lrelu(p[i]);
}

// ---------------------------------------------------------------------------
// nnHyper[r,:] = (r<nu ? uE[r,:] @ uH : iE[r-nu,:] @ iH)   [ntot,128]x[128,128]
// 8 waves/block, wave computes a 16x128 output slab via 8 accumulators.
// ---------------------------------------------------------------------------
__global__ __launch_bounds__(256) void k_hyper_gemm(const float* __restrict__ uE,
                                                    const float* __restrict__ iE,
                                                    const float* __restrict__ uH,
                                                    const float* __restrict__ iH,
                                                    float* __restrict__ out,
                                                    int nu, int ntot) {
  const int lane = threadIdx.x & 31;
  const int wave = threadIdx.x >> 5;
  const int half = lane >> 4;
  const int lr = lane & 15;
  const int m0 = blockIdx.x * 128 + wave * 16;
  if (m0 >= ntot) return;  // whole-wave exit: EXEC stays all-ones for WMMA
  const float* A = (m0 < nu) ? uE + (size_t)m0 * DD : iE + (size_t)(m0 - nu) * DD;
  const float* B = (m0 < nu) ? uH : iH;

  v8f acc[8];
#pragma unroll
  for (int t = 0; t < 8; ++t) acc[t] = (v8f)0.0f;

  for (int k = 0; k < DD; k += 4) {
    const float* ap = A + (size_t)lr * DD + k + 2 * half;
    v2f a;
    a.x = ap[0];
    a.y = ap[1];
    const float* bp = B + (size_t)(k + 2 * half) * DD + lr;
#pragma unroll
    for (int nt = 0; nt < 8; ++nt) {
      v2f b;
      b.x = bp[nt * 16];
      b.y = bp[DD + nt * 16];
      acc[nt] = wmma4(a, b, acc[nt]);
    }
  }
  float* o = out + (size_t)m0 * DD;
#pragma unroll
  for (int nt = 0; nt < 8; ++nt)
#pragma unroll
    for (int g = 0; g < 8; ++g)
      o[(size_t)(g + 8 * half) * DD + nt * 16 + lr] = acc[nt][g];
}

// ---------------------------------------------------------------------------
// latSeg += adjH_seg^T @ prev_seg   (partial over a row-chunk, fp32 atomics)
// blockIdx.y: 0 = users, 1 = items. A[m][k] = adjH[row k][col m].
// ---------------------------------------------------------------------------
__global__ __launch_bounds__(256) void k_latT(const float* __restrict__ nnHyper,
                                              const float* __restrict__ prev,
                                              float* __restrict__ latU,
                                              float* __restrict__ latI,
                                              int nu, int ni, int chunk) {
  const int seg = blockIdx.y;
  const int segRows = seg ? ni : nu;
  const int r0 = blockIdx.x * chunk;
  if (r0 >= segRows) return;
  const int kcnt = min(chunk, segRows - r0);
  const size_t segOff = seg ? (size_t)nu * DD : 0;
  const float* Ah = nnHyper + segOff + (size_t)r0 * DD;
  const float* Pv = prev + segOff + (size_t)r0 * DD;
  float* out = seg ? latI : latU;

  const int lane = threadIdx.x & 31;
  const int wave = threadIdx.x >> 5;
  const int half = lane >> 4;
  const int lr = lane & 15;
  const int m0 = wave * 16;

  v8f acc[8];
#pragma unroll
  for (int t = 0; t < 8; ++t) acc[t] = (v8f)0.0f;

  for (int k = 0; k + 4 <= kcnt; k += 4) {
    const float* ap = Ah + (size_t)(k + 2 * half) * DD + m0 + lr;
    v2f a;
    a.x = ap[0];
    a.y = ap[DD];
    const float* bp = Pv + (size_t)(k + 2 * half) * DD + lr;
#pragma unroll
    for (int nt = 0; nt < 8; ++nt) {
      v2f b;
      b.x = bp[nt * 16];
      b.y = bp[DD + nt * 16];
      acc[nt] = wmma4(a, b, acc[nt]);
    }
  }
#pragma unroll
  for (int nt = 0; nt < 8; ++nt)
#pragma unroll
    for (int g = 0; g < 8; ++g)
      atomicAdd(&out[(size_t)(m0 + g + 8 * half) * DD + nt * 16 + lr], acc[nt][g]);
}

// ---------------------------------------------------------------------------
// latOut = leaky(V @ latIn) + latIn     (one 128x128x128 GEMM per block)
// grid.x = 2 selects the U / I matrix.
// ---------------------------------------------------------------------------
__global__ __launch_bounds__(256) void k_vloop(const float* __restrict__ V,
                                               const float* __restrict__ latIn,
                                               float* __restrict__ latOut) {
  const float* Bi = latIn + (size_t)blockIdx.x * DD * DD;
  float* Bo = latOut + (size_t)blockIdx.x * DD * DD;
  const int lane = threadIdx.x & 31;
  const int wave = threadIdx.x >> 5;
  const int half = lane >> 4;
  const int lr = lane & 15;
  const int m0 = wave * 16;

  v8f acc[8];
#pragma unroll
  for (int t = 0; t < 8; ++t) acc[t] = (v8f)0.0f;

  for (int k = 0; k < DD; k += 4) {
    const float* ap = V + (size_t)(m0 + lr) * DD + k + 2 * half;
    v2f a;
    a.x = ap[0];
    a.y = ap[1];
    const float* bp = Bi + (size_t)(k + 2 * half) * DD + lr;
#pragma unroll
    for (int nt = 0; nt < 8; ++nt) {
      v2f b;
      b.x = bp[nt * 16];
      b.y = bp[DD + nt * 16];
      acc[nt] = wmma4(a, b, acc[nt]);
    }
  }
#pragma unroll
  for (int nt = 0; nt < 8; ++nt)
#pragma unroll
    for (int g = 0; g < 8; ++g) {
      size_t idx = (size_t)(m0 + g + 8 * half) * DD + nt * 16 + lr;
      Bo[idx] = lrelu(acc[nt][g]) + Bi[idx];
    }
}

// ---------------------------------------------------------------------------
// hyp = leaky(nnHyper @ lat_{U or I})   (per-tile lat selection by segment)
// ---------------------------------------------------------------------------
__global__ __launch_bounds__(256) void k_apply_lat(const float* __restrict__ nnHyper,
                                                   const float* __restrict__ latU,
                                                   const float* __restrict__ latI,
                                                   float* __restrict__ out,
                                                   int nu, int ntot) {
  const int lane = threadIdx.x & 31;
  const int wave = threadIdx.x >> 5;
  const int half = lane >> 4;
  const int lr = lane & 15;
  const int m0 = blockIdx.x * 128 + wave * 16;
  if (m0 >= ntot) return;
  const float* A = nnHyper + (size_t)m0 * DD;
  const float* B = (m0 < nu) ? latU : latI;

  v8f acc[8];
#pragma unroll
  for (int t = 0; t < 8; ++t) acc[t] = (v8f)0.0f;

  for (int k = 0; k < DD; k += 4) {
    const float* ap = A + (size_t)lr * DD + k + 2 * half;
    v2f a;
    a.x = ap[0];
    a.y = ap[1];
    const float* bp = B + (size_t)(k + 2 * half) * DD + lr;
#pragma unroll
    for (int nt = 0; nt < 8; ++nt) {
      v2f b;
      b.x = bp[nt * 16];
      b.y = bp[DD + nt * 16];
      acc[nt] = wmma4(a, b, acc[nt]);
    }
  }
  float* o = out + (size_t)m0 * DD;
#pragma unroll
  for (int nt = 0; nt < 8; ++nt)
#pragma unroll
    for (int g = 0; g < 8; ++g)
      o[(size_t)(g + 8 * half) * DD + nt * 16 + lr] = lrelu(acc[nt][g]);
}

// ---------------------------------------------------------------------------
// SpMM scatter: buf[row,:] += val * prev[col,:]  (32 threads/edge, float4 each)
// ---------------------------------------------------------------------------
__global__ __launch_bounds__(256) void k_spmm(const int* __restrict__ rows,
                                              const int* __restrict__ cols,
                                              const float* __restrict__ vals,
                                              const float* __restrict__ prev,
                                              float* __restrict__ buf, int nE) {
  long tid = (long)blockIdx.x * blockDim.x + threadIdx.x;
  long e = tid >> 5;
  if (e >= nE) return;
  int sub = (int)(tid & 31) * 4;
  int r = rows[e];
  int c = cols[e];
  float v = vals[e];
  const float4 g = *(const float4*)(prev + (size_t)c * DD + sub);
  float* o = buf + (size_t)r * DD + sub;
  atomicAdd(o + 0, v * g.x);
  atomicAdd(o + 1, v * g.y);
  atomicAdd(o + 2, v * g.z);
  atomicAdd(o + 3, v * g.w);
}

// ---------------------------------------------------------------------------
// tem = leaky(buf); gnn = tem; next = tem + hyp; prev = next; acc += next
// ---------------------------------------------------------------------------
__global__ __launch_bounds__(256) void k_combine(const float* __restrict__ buf,
                                                 const float* __restrict__ hyp,
                                                 float* __restrict__ gnn,
                                                 float* __restrict__ prev,
                                                 float* __restrict__ acc, long tot) {
  long i = (long)blockIdx.x * blockDim.x + threadIdx.x;
  if (i >= tot) return;
  float t = lrelu(buf[i]);
  gnn[i] = t;
  float nx = t + hyp[i];
  prev[i] = nx;
  acc[i] += nx;
}

// ---------------------------------------------------------------------------
extern "C" void kernel_launch(void* const* d_in, const int* in_sizes, int n_in,
                              void* d_out, int out_size, void* d_ws, size_t ws_size,
                              hipStream_t stream) {
  const float* uE = (const float*)d_in[0];
  const float* iE = (const float*)d_in[1];
  const float* uH = (const float*)d_in[2];
  const float* iH = (const float*)d_in[3];
  const float* V = (const float*)d_in[4];
  const int* rows = (const int*)d_in[5];
  const int* cols = (const int*)d_in[6];
  const float* vals = (const float*)d_in[7];

  const int nu = in_sizes[0] / DD;
  const int ni = in_sizes[1] / DD;
  const int ntot = nu + ni;
  const int nE = in_sizes[5];
  const long tot = (long)ntot * DD;  // N*D elements

  float* out = (float*)d_out;
  float* embeds = out;                  // [N,D]
  float* gnn = out + tot;               // [K,N,D]
  float* hypO = out + tot + 2 * tot;    // [K,N,D]

  // Workspace layout (contiguous): nnHyper | prev | spmm_buf | latA(2) | latB(2)
  float* nnH = (float*)d_ws;
  float* prev = nnH + tot;
  float* buf = prev + tot;
  float* latA = buf + tot;              // latA: U then I (2*128*128)
  float* latB = latA + 2 * DD * DD;

  const dim3 blk(256);
  const int gElem = (int)((tot + 255) / 256);
  const int gGemm = (ntot + 127) / 128;
  const int CHUNK = 1024;
  const int gLat = (max(nu, ni) + CHUNK - 1) / CHUNK;

  k_init<<<gElem, blk, 0, stream>>>(uE, iE, prev, embeds, nu, tot);
  k_hyper_gemm<<<gGemm, blk, 0, stream>>>(uE, iE, uH, iH, nnH, nu, ntot);

  for (int layer = 0; layer < 2; ++layer) {
    // zero spmm buffer + lat accumulators (contiguous region)
    const long zc = tot + 2 * DD * DD;
    k_fill<<<(int)((zc + 255) / 256), blk, 0, stream>>>(buf, zc);

    // GCN path: scatter-add SpMM
    const long sThreads = (long)nE * 32;
    k_spmm<<<(int)((sThreads + 255) / 256), blk, 0, stream>>>(rows, cols, vals,
                                                              prev, buf, nE);

    // Hyper path: lat = leaky(adjH^T @ prev); C x (lat = leaky(V@lat)+lat)
    k_latT<<<dim3(gLat, 2), blk, 0, stream>>>(nnH, prev, latA, latA + DD * DD,
                                              nu, ni, CHUNK);
    k_leaky_ip<<<(2 * DD * DD + 255) / 256, blk, 0, stream>>>(latA, 2 * DD * DD);
    k_vloop<<<2, blk, 0, stream>>>(V, latA, latB);
    k_vloop<<<2, blk, 0, stream>>>(V, latB, latA);

    // hyp = leaky(adjH @ lat) -> directly into hyper_lats[layer]
    float* hypSlab = hypO + (size_t)layer * tot;
    k_apply_lat<<<gGemm, blk, 0, stream>>>(nnH, latA, latA + DD * DD, hypSlab,
                                           nu, ntot);

    // tem = leaky(buf); emit gnn_lats[layer]; prev = tem + hyp; embeds += prev
    float* gnnSlab = gnn + (size_t)layer * tot;
    k_combine<<<gElem, blk, 0, stream>>>(buf, hypSlab, gnnSlab, prev, embeds, tot);
  }
}